// Conv_Transformer_52639119179874
// MI455X (gfx1250) — compile-verified
//
#include <hip/hip_runtime.h>
#include <hip/hip_bf16.h>

// Problem constants (match reference)
#define Bb 4
#define Cc 64
#define Hh 64
#define Ww 64
#define Ss 20
#define HWS (Hh * Ww * Ss)   // 81920 positions per (b, channel)
#define OC  (3 * Cc)         // 192 qkv channels
#define NPOS (Bb * Hh * Ww * Ss)   // 327680
#define NBHW (Bb * Hh * Ww)        // 16384

typedef __attribute__((ext_vector_type(2))) float v2f;
typedef __attribute__((ext_vector_type(8))) float v8f;

// ---------------------------------------------------------------------------
// Kernel 1: qkv = W1(192x64) @ X_b(64x81920) + b1, per batch, via WMMA f32.
// Grid: (81920/(8*64), 192/16, B)  Block: 256 threads = 8 waves.
// Each wave computes a 16(M) x 64(N) output strip with 4 f32 accumulators,
// reusing the A (weight) fragment across the 4 N-subtiles.
// ---------------------------------------------------------------------------
__global__ __launch_bounds__(256) void qkv_gemm_wmma(
    const float* __restrict__ x,    // (B, C, H, W, S)
    const float* __restrict__ w1,   // (192, 64)
    const float* __restrict__ b1,   // (192,)
    float* __restrict__ qkv)        // (B, 192, H, W, S)
{
    const int lane = threadIdx.x & 31;
    const int wave = threadIdx.x >> 5;
    const int j  = lane & 15;   // N / M index within tile
    const int h2 = lane >> 4;   // lane-half selects K pair / M upper half
    const int n0 = (blockIdx.x * 8 + wave) * 64;  // position tile base
    const int m0 = blockIdx.y * 16;               // output-channel tile base
    const int b  = blockIdx.z;

    const float* xb = x + (size_t)b * Cc * HWS;

    v8f acc[4] = {v8f{0.f}, v8f{0.f}, v8f{0.f}, v8f{0.f}};

    for (int k0 = 0; k0 < Cc; k0 += 4) {
        // A fragment (16x4 f32): lane(j,h2) holds W1[m0+j][k0+2*h2 + {0,1}]
        const float* wrow = w1 + (size_t)(m0 + j) * Cc + (k0 + 2 * h2);
        v2f afrag;
        afrag.x = wrow[0];
        afrag.y = wrow[1];
#pragma unroll
        for (int nt = 0; nt < 4; ++nt) {
            // B fragment (4x16 f32): lane(j,h2) holds X[k0+2*h2 + {0,1}][p]
            const float* xcol = xb + (size_t)(k0 + 2 * h2) * HWS + (n0 + nt * 16 + j);
            v2f bfrag;
            bfrag.x = xcol[0];
            bfrag.y = xcol[HWS];
            acc[nt] = __builtin_amdgcn_wmma_f32_16x16x4_f32(
                false, afrag, false, bfrag, (short)0, acc[nt], false, false);
        }
    }

    // D layout: VGPR r, lane(j,h2) -> D[M = r + 8*h2][N = j]
#pragma unroll
    for (int nt = 0; nt < 4; ++nt) {
        const int p = n0 + nt * 16 + j;
#pragma unroll
        for (int r = 0; r < 8; ++r) {
            const int ch = m0 + r + 8 * h2;
            qkv[((size_t)b * OC + ch) * HWS + p] = acc[nt][r] + b1[ch];
        }
    }
}

// ---------------------------------------------------------------------------
// Kernel 2: single-output-channel 3x3 SAME convs:
//   A  = conv(Q, w2[:, 0:64]),  Bk = conv(K, w2[:, 64:128])
// One thread per (b,h,w,s); s is innermost -> coalesced 20*4B runs; the 9x
// spatial re-reads of Q/K hit L2 (Q and K each 84MB, L2 = 192MB).
// Weight indices are wave-uniform -> scalar loads.
// ---------------------------------------------------------------------------
__global__ __launch_bounds__(256) void qk_conv3x3(
    const float* __restrict__ qkv,  // (B, 192, H, W, S)
    const float* __restrict__ w2,   // (1, 128, 3, 3)
    float* __restrict__ Aout,       // (B, H, W, S) flat
    float* __restrict__ Bkout)      // (B, H, W, S) flat
{
    const int idx = blockIdx.x * blockDim.x + threadIdx.x;
    if (idx >= NPOS) return;
    const int s = idx % Ss;
    const int w = (idx / Ss) % Ww;
    const int h = (idx / (Ss * Ww)) % Hh;
    const int b = idx / (Ss * Ww * Hh);

    const float* Qb = qkv + (size_t)b * OC * HWS;           // channels 0..63
    const float* Kb = Qb + (size_t)Cc * HWS;                // channels 64..127

    float accA = 0.f, accB = 0.f;
    for (int c = 0; c < Cc; ++c) {
        const float* qc = Qb + (size_t)c * HWS;
        const float* kc = Kb + (size_t)c * HWS;
        const float* wq = w2 + c * 9;             // uniform -> SGPR loads
        const float* wk = w2 + (Cc + c) * 9;
#pragma unroll
        for (int dy = 0; dy < 3; ++dy) {
            const int hh = h + dy - 1;
            if (hh < 0 || hh >= Hh) continue;
#pragma unroll
            for (int dx = 0; dx < 3; ++dx) {
                const int ww = w + dx - 1;
                if (ww < 0 || ww >= Ww) continue;
                const int off = (hh * Ww + ww) * Ss + s;
                accA = fmaf(qc[off], wq[dy * 3 + dx], accA);
                accB = fmaf(kc[off], wk[dy * 3 + dx], accB);
            }
        }
    }
    Aout[idx]  = accA;
    Bkout[idx] = accB;
}

// ---------------------------------------------------------------------------
// Kernel 3: faithful softmax over t of (A[s]+Bk[t]+b2), then
//   out[b,c,h,w,s] = sum_t attn[t][s] * V[b,c,h,w,t]
// Block = 256 threads = 4 groups x 64 channels; each group owns one (b,h,w).
// Threads c<20 of a group build the 20x20 attn matrix in LDS (one s each).
// V[...,t] is contiguous in t; out[...,s] contiguous in s -> coalesced.
// ---------------------------------------------------------------------------
__global__ __launch_bounds__(256) void attn_apply(
    const float* __restrict__ qkv,   // V = channels 128..191
    const float* __restrict__ Aout,
    const float* __restrict__ Bkout,
    const float* __restrict__ b2,    // (1,)
    float* __restrict__ out)         // (B, C, H, W, S)
{
    __shared__ float attn[4][Ss * Ss];   // [group][s*Ss + t]

    const int g = threadIdx.x >> 6;      // 0..3
    const int c = threadIdx.x & 63;      // 0..63
    const int pos = blockIdx.x * 4 + g;  // over B*H*W = 16384
    const int w = pos % Ww;
    const int h = (pos / Ww) % Hh;
    const int b = pos / (Ww * Hh);

    const float* Arow = Aout  + (size_t)pos * Ss;
    const float* Brow = Bkout + (size_t)pos * Ss;
    const float bias = b2[0];

    if (c < Ss) {
        const int s = c;
        const float As = Arow[s];
        float sc[Ss];
        float m = -3.402823466e+38f;
#pragma unroll
        for (int t = 0; t < Ss; ++t) {
            sc[t] = (As + Brow[t]) + bias;       // match reference add order
            m = fmaxf(m, sc[t]);
        }
        float e[Ss];
        float denom = 0.f;
#pragma unroll
        for (int t = 0; t < Ss; ++t) {
            e[t] = __expf(sc[t] - m);
            denom += e[t];
        }
#pragma unroll
        for (int t = 0; t < Ss; ++t)
            attn[g][s * Ss + t] = e[t] / denom;
    }
    __syncthreads();

    // V row (contiguous in t)
    const float* vrow = qkv + ((size_t)b * OC + 2 * Cc + c) * HWS
                            + (size_t)(h * Ww + w) * Ss;
    float v[Ss];
#pragma unroll
    for (int t = 0; t < Ss; ++t) v[t] = vrow[t];

    float* orow = out + ((size_t)b * Cc + c) * (size_t)(Hh * Ww) * Ss
                      + (size_t)(h * Ww + w) * Ss;
#pragma unroll
    for (int s = 0; s < Ss; ++s) {
        float acc = 0.f;
#pragma unroll
        for (int t = 0; t < Ss; ++t)
            acc = fmaf(v[t], attn[g][s * Ss + t], acc);
        orow[s] = acc;
    }
}

// ---------------------------------------------------------------------------
// Launcher. ws layout: [ qkv : B*192*HWS f32 ][ A : NPOS f32 ][ Bk : NPOS f32 ]
// ---------------------------------------------------------------------------
extern "C" void kernel_launch(void* const* d_in, const int* in_sizes, int n_in,
                              void* d_out, int out_size, void* d_ws, size_t ws_size,
                              hipStream_t stream) {
    const float* x  = (const float*)d_in[0];
    const float* w1 = (const float*)d_in[1];
    const float* b1 = (const float*)d_in[2];
    const float* w2 = (const float*)d_in[3];
    const float* b2 = (const float*)d_in[4];
    float* out = (float*)d_out;

    float* qkv = (float*)d_ws;
    float* A   = qkv + (size_t)Bb * OC * HWS;
    float* Bk  = A + NPOS;

    // Kernel 1: WMMA GEMM. N chunks: 81920 / (8 waves * 64) = 160.
    dim3 g1(HWS / (8 * 64), OC / 16, Bb);
    qkv_gemm_wmma<<<g1, 256, 0, stream>>>(x, w1, b1, qkv);

    // Kernel 2: 3x3 convs over Q, K.
    qk_conv3x3<<<(NPOS + 255) / 256, 256, 0, stream>>>(qkv, w2, A, Bk);

    // Kernel 3: softmax + apply to V.
    attn_apply<<<NBHW / 4, 256, 0, stream>>>(qkv, A, Bk, b2, out);
}